// GraphEncoder_6846177870029
// MI455X (gfx1250) — compile-verified
//
#include <hip/hip_runtime.h>

// GraphEncoder for MI455X (gfx1250): fp32 WMMA GEMMs + fused edge gather/scatter.
// All GEMMs use V_WMMA_F32_16X16X4_F32 (wave32, 16x16 C/D tile per wave).

typedef __attribute__((ext_vector_type(2))) float v2f;
typedef __attribute__((ext_vector_type(8))) float v8f;

__device__ __forceinline__ v8f wmma4(v2f a, v2f b, v8f c) {
  // 8 args: (neg_a, A, neg_b, B, c_mod, C, reuse_a, reuse_b)
  return __builtin_amdgcn_wmma_f32_16x16x4_f32(false, a, false, b, (short)0, c,
                                               false, false);
}

// ---------------------------------------------------------------------------
// out[M,64] = act(A[M,64] @ W[64,64] + bias[64]) ; one block = 16 rows x 64
// cols (4 waves, each a 16x16 tile). W and the A tile staged through LDS.
// ---------------------------------------------------------------------------
__global__ __launch_bounds__(128) void gemm64_kernel(
    const float* __restrict__ A, const float* __restrict__ W,
    const float* __restrict__ bias, float* __restrict__ out, int relu) {
  __shared__ float ldsW[64 * 64];
  __shared__ float ldsA[16 * 64];
  const int r0 = blockIdx.x * 16;
  for (int i = threadIdx.x; i < 64 * 64; i += 128) ldsW[i] = W[i];
  for (int i = threadIdx.x; i < 16 * 64; i += 128) ldsA[i] = A[r0 * 64 + i];
  __syncthreads();

  const int lane = threadIdx.x & 31;
  const int wv = threadIdx.x >> 5;   // column tile 0..3
  const int half = lane >> 4;        // 0 or 1
  const int mr = lane & 15;          // A row within tile / B col within tile
  const int n = wv * 16 + mr;        // output column

  v8f c;
  const float bn = bias[n];
#pragma unroll
  for (int i = 0; i < 8; ++i) c[i] = bn;

#pragma unroll
  for (int k0 = 0; k0 < 64; k0 += 4) {
    // A 16x4 f32 fragment: k = vgpr + 2*half
    v2f a = *(const v2f*)&ldsA[mr * 64 + k0 + 2 * half];
    v2f b;
    b.x = ldsW[(k0 + 2 * half) * 64 + n];
    b.y = ldsW[(k0 + 2 * half + 1) * 64 + n];
    c = wmma4(a, b, c);
  }

#pragma unroll
  for (int i = 0; i < 8; ++i) {
    float v = c[i];
    if (relu) v = fmaxf(v, 0.0f);
    out[(r0 + i + 8 * half) * 64 + n] = v;  // C/D layout: M = i + 8*half
  }
}

// ---------------------------------------------------------------------------
// Edge phase, fully fused:
//   ef  = edge_attr[E,16] @ edge_W[16,64] + edge_b        (WMMA, K=16)
//   msg = hn[src] + ef                                     (L2 gather)
//   aggr[dst] += msg                                       (L2 atomic scatter)
// One block = 16 edges x 64 cols.
// ---------------------------------------------------------------------------
__global__ __launch_bounds__(128) void edge_fused_kernel(
    const float* __restrict__ EA, const float* __restrict__ W,
    const float* __restrict__ bias, const int* __restrict__ src,
    const int* __restrict__ dst, const float* __restrict__ hn,
    float* __restrict__ aggr) {
  __shared__ float ldsW[16 * 64];
  const int e0 = blockIdx.x * 16;
  for (int i = threadIdx.x; i < 16 * 64; i += 128) ldsW[i] = W[i];
  __syncthreads();

  const int lane = threadIdx.x & 31;
  const int wv = threadIdx.x >> 5;
  const int half = lane >> 4;
  const int mr = lane & 15;
  const int n = wv * 16 + mr;

  v8f c;
  const float bn = bias[n];
#pragma unroll
  for (int i = 0; i < 8; ++i) c[i] = bn;

#pragma unroll
  for (int k0 = 0; k0 < 16; k0 += 4) {
    v2f a = *(const v2f*)&EA[(size_t)(e0 + mr) * 16 + k0 + 2 * half];
    v2f b;
    b.x = ldsW[(k0 + 2 * half) * 64 + n];
    b.y = ldsW[(k0 + 2 * half + 1) * 64 + n];
    c = wmma4(a, b, c);
  }

#pragma unroll
  for (int i = 0; i < 8; ++i) {
    const int e = e0 + i + 8 * half;
    const int s = src[e];
    const int d = dst[e];
    const float msg = c[i] + hn[(size_t)s * 64 + n];
    atomicAdd(&aggr[(size_t)d * 64 + n], msg);
  }
}

// ---------------------------------------------------------------------------
// out[M,64] = relu([A0|A1][M,128] @ W[128,64] + bias)   (mlp1 on concat)
// ---------------------------------------------------------------------------
__global__ __launch_bounds__(128) void gemm_cat128_kernel(
    const float* __restrict__ A0, const float* __restrict__ A1,
    const float* __restrict__ W, const float* __restrict__ bias,
    float* __restrict__ out) {
  __shared__ float ldsW[128 * 64];
  __shared__ float ldsA[16 * 128];
  const int r0 = blockIdx.x * 16;
  for (int i = threadIdx.x; i < 128 * 64; i += 128) ldsW[i] = W[i];
  for (int i = threadIdx.x; i < 16 * 64; i += 128) {
    const int row = i >> 6, col = i & 63;
    ldsA[row * 128 + col] = A0[r0 * 64 + i];
    ldsA[row * 128 + 64 + col] = A1[r0 * 64 + i];
  }
  __syncthreads();

  const int lane = threadIdx.x & 31;
  const int wv = threadIdx.x >> 5;
  const int half = lane >> 4;
  const int mr = lane & 15;
  const int n = wv * 16 + mr;

  v8f c;
  const float bn = bias[n];
#pragma unroll
  for (int i = 0; i < 8; ++i) c[i] = bn;

#pragma unroll
  for (int k0 = 0; k0 < 128; k0 += 4) {
    v2f a = *(const v2f*)&ldsA[mr * 128 + k0 + 2 * half];
    v2f b;
    b.x = ldsW[(k0 + 2 * half) * 64 + n];
    b.y = ldsW[(k0 + 2 * half + 1) * 64 + n];
    c = wmma4(a, b, c);
  }

#pragma unroll
  for (int i = 0; i < 8; ++i)
    out[(r0 + i + 8 * half) * 64 + n] = fmaxf(c[i], 0.0f);
}

// ---------------------------------------------------------------------------
// GraphNorm helpers
// ---------------------------------------------------------------------------
__global__ void fill_zero_kernel(float* __restrict__ p, int n) {
  const int i = blockIdx.x * blockDim.x + threadIdx.x;
  if (i < n) p[i] = 0.0f;
}

__global__ void stats_kernel(const float* __restrict__ h,
                             const int* __restrict__ batch,
                             float* __restrict__ sum, float* __restrict__ sum2,
                             float* __restrict__ cnt, int total) {
  const int idx = blockIdx.x * blockDim.x + threadIdx.x;
  if (idx >= total) return;
  const int node = idx >> 6;
  const int col = idx & 63;
  const float v = h[idx];
  const int g = batch[node];
  atomicAdd(&sum[g * 64 + col], v);
  atomicAdd(&sum2[g * 64 + col], v * v);
  if (col == 0) atomicAdd(&cnt[g], 1.0f);
}

__global__ void norm_apply_kernel(
    const float* __restrict__ h2, const float* __restrict__ hres,
    const int* __restrict__ batch, const float* __restrict__ sum,
    const float* __restrict__ sum2, const float* __restrict__ cnt,
    const float* __restrict__ w, const float* __restrict__ b,
    float* __restrict__ out, int total) {
  const int idx = blockIdx.x * blockDim.x + threadIdx.x;
  if (idx >= total) return;
  const int node = idx >> 6;
  const int col = idx & 63;
  const int g = batch[node];
  const float c = fmaxf(cnt[g], 1.0f);
  const float m = sum[g * 64 + col] / c;
  const float m2 = sum2[g * 64 + col] / c;
  const float var = m2 - m * m;
  const float x = h2[idx];
  float y = w[col] * ((x - m) * rsqrtf(var + 1e-5f)) + b[col];
  y = fmaxf(y, 0.0f) + hres[idx];  // relu then residual
  out[idx] = y;
}

// ---------------------------------------------------------------------------
// Host driver
// ---------------------------------------------------------------------------
extern "C" void kernel_launch(void* const* d_in, const int* in_sizes, int n_in,
                              void* d_out, int out_size, void* d_ws,
                              size_t ws_size, hipStream_t stream) {
  const float* x = (const float*)d_in[0];
  const int* ei = (const int*)d_in[1];       // [2,E]
  const float* ea = (const float*)d_in[2];   // [E,16]
  const int* batch = (const int*)d_in[3];    // [N]
  const float* enc_W = (const float*)d_in[4];
  const float* enc_b = (const float*)d_in[5];
  const float* node_W = (const float*)d_in[6];
  const float* node_b = (const float*)d_in[7];
  const float* edge_W = (const float*)d_in[8];
  const float* edge_b = (const float*)d_in[9];
  const float* mlp1_W = (const float*)d_in[10];
  const float* mlp1_b = (const float*)d_in[11];
  const float* mlp2_W = (const float*)d_in[12];
  const float* mlp2_b = (const float*)d_in[13];
  const float* norm_w = (const float*)d_in[14];
  const float* norm_b = (const float*)d_in[15];
  float* out = (float*)d_out;

  const int N = in_sizes[0] / 64;            // 100000
  const int E = in_sizes[2] / 16;            // 1200000
  const int G = 1000;
  const int NH = N * 64;
  const int Lnum = 3;

  float* ws = (float*)d_ws;
  float* bufA = ws;             // h / h_res
  float* bufB = ws + (size_t)NH;        // hn, later h2
  float* bufC = ws + (size_t)2 * NH;    // aggr
  float* bufD = ws + (size_t)3 * NH;    // t (post-mlp1)
  float* sum = ws + (size_t)4 * NH;     // [G,64]
  float* sum2 = sum + (size_t)G * 64;   // [G,64]
  float* cnt = sum2 + (size_t)G * 64;   // [G]
  const int statsN = 2 * G * 64 + G;

  const int* src = ei;
  const int* dst = ei + E;

  // encoder: h = x @ enc_W + enc_b
  gemm64_kernel<<<N / 16, 128, 0, stream>>>(x, enc_W, enc_b, bufA, 0);

  for (int l = 0; l < Lnum; ++l) {
    // hn = h @ node_W[l] + node_b[l]
    gemm64_kernel<<<N / 16, 128, 0, stream>>>(bufA, node_W + l * 64 * 64,
                                              node_b + l * 64, bufB, 0);
    // zero aggr + stats
    fill_zero_kernel<<<(NH + 255) / 256, 256, 0, stream>>>(bufC, NH);
    fill_zero_kernel<<<(statsN + 255) / 256, 256, 0, stream>>>(sum, statsN);
    // ef = EA @ edge_W + b ; msg = hn[src] + ef ; aggr[dst] += msg
    edge_fused_kernel<<<E / 16, 128, 0, stream>>>(
        ea, edge_W + l * 16 * 64, edge_b + l * 64, src, dst, bufB, bufC);
    // t = relu([hn|aggr] @ mlp1_W + b)
    gemm_cat128_kernel<<<N / 16, 128, 0, stream>>>(
        bufB, bufC, mlp1_W + l * 128 * 64, mlp1_b + l * 64, bufD);
    // h2 = t @ mlp2_W + b   (reuse bufB)
    gemm64_kernel<<<N / 16, 128, 0, stream>>>(bufD, mlp2_W + l * 64 * 64,
                                              mlp2_b + l * 64, bufB, 0);
    // graph norm + relu + residual
    stats_kernel<<<(NH + 255) / 256, 256, 0, stream>>>(bufB, batch, sum, sum2,
                                                       cnt, NH);
    float* dsth = (l == Lnum - 1) ? out : bufA;
    norm_apply_kernel<<<(NH + 255) / 256, 256, 0, stream>>>(
        bufB, bufA, batch, sum, sum2, cnt, norm_w + l * 64, norm_b + l * 64,
        dsth, NH);
  }
}